// GaussianModel2D_46892452938269
// MI455X (gfx1250) — compile-verified
//
#include <hip/hip_runtime.h>
#include <hip/hip_bf16.h>

typedef __attribute__((ext_vector_type(2))) float v2f;
typedef __attribute__((ext_vector_type(8))) float v8f;

namespace {
constexpr int kH = 512;
constexpr int kW = 512;
constexpr int kN = 256;               // gaussians; also blockDim.x
constexpr int kPixPerWave = 16;       // WMMA M dimension
constexpr int kWavesPerBlock = 8;     // 256 threads / wave32
constexpr int kPixPerBlock = kPixPerWave * kWavesPerBlock;  // 128
}

// One wave renders 16 pixels. Both half-waves (lanes 0-15 / 16-31) run the
// sequential front-to-back compositing scan for the SAME 16 pixels so that the
// f32 WMMA A-operand layout (K={0,1} in lanes 0-15, K={2,3} in lanes 16-31)
// can be fed without cross-lane shuffles. Color reduction (weights x colors)
// runs on V_WMMA_F32_16X16X4_F32, accumulated over 64 K-chunks of 4.
//
// B-operand trick: D columns 3..15 are never read out, so B columns >= 3 need
// NOT be zeroed -- the channel index is clamped (min(m,2)) and the two colors
// are fetched with one unconditional 8-byte LDS load, keeping the inner loop
// free of divergent control flow / exec-mask saves.
__global__ __launch_bounds__(256) void gs2d_render_kernel(
    const float* __restrict__ coords,       // (H,W,2)
    const float* __restrict__ means,        // (N,2)
    const float* __restrict__ log_scales,   // (N,2)
    const float* __restrict__ rotations,    // (N,)
    const float* __restrict__ raw_colors,   // (N,3)
    const float* __restrict__ raw_opac,     // (N,)
    float* __restrict__ out)                // (H,W,3)
{
    __shared__ __align__(16) float s_a[kN], s_b2[kN], s_c[kN];
    __shared__ __align__(16) float s_mx[kN], s_my[kN], s_op[kN];
    __shared__ __align__(16) float s_col[3][kN];     // [channel][gaussian]
    __shared__ float s_T[kPixPerBlock];              // final transmittance (bg)
    __shared__ float s_out[kPixPerBlock * 3];

    const int tid = threadIdx.x;

    // ---- per-gaussian precompute (redundant per block; one gaussian/thread) ----
    {
        const int n = tid;
        const float lsx = log_scales[2 * n + 0];
        const float lsy = log_scales[2 * n + 1];
        const float i2sx = 0.5f * __expf(-2.0f * lsx);   // 1 / (2*sx^2)
        const float i2sy = 0.5f * __expf(-2.0f * lsy);   // 1 / (2*sy^2)
        float sr, cr;
        __sincosf(rotations[n], &sr, &cr);
        const float cr2 = cr * cr, sr2 = sr * sr;
        s_a[n]  = cr2 * i2sx + sr2 * i2sy;
        s_b2[n] = 2.0f * (sr * cr) * (i2sy - i2sx);      // 2*b
        s_c[n]  = sr2 * i2sx + cr2 * i2sy;
        s_mx[n] = means[2 * n + 0];
        s_my[n] = means[2 * n + 1];
        s_op[n] = 1.0f / (1.0f + __expf(-raw_opac[n]));
        s_col[0][n] = 1.0f / (1.0f + __expf(-raw_colors[3 * n + 0]));
        s_col[1][n] = 1.0f / (1.0f + __expf(-raw_colors[3 * n + 1]));
        s_col[2][n] = 1.0f / (1.0f + __expf(-raw_colors[3 * n + 2]));
    }
    __syncthreads();

    const int lane = tid & 31;
    const int wave = tid >> 5;
    const int m    = lane & 15;       // pixel row within the wave's 16-pixel tile
    const int hi   = lane >> 4;       // 0 -> K pair {0,1}, 1 -> K pair {2,3}
    const int pix  = blockIdx.x * kPixPerBlock + wave * kPixPerWave + m;

    const float px = coords[2 * pix + 0];
    const float py = coords[2 * pix + 1];

    const int  n_col = m;                        // B column = color channel
    const int  cch   = (n_col < 3) ? n_col : 2;  // clamped channel (cols >=3 are junk, never read)
    const float* colrow = &s_col[cch][0];

    float T = 1.0f;                   // running transmittance for pixel m
    v8f acc = {0.f, 0.f, 0.f, 0.f, 0.f, 0.f, 0.f, 0.f};

    #pragma unroll 4
    for (int k0 = 0; k0 < kN; k0 += 4) {
        float w0, w1, w2, w3;
        {
            float dx, dy, e, alpha;
            dx = px - s_mx[k0 + 0]; dy = py - s_my[k0 + 0];
            e = -(s_a[k0 + 0] * dx * dx + s_b2[k0 + 0] * dx * dy + s_c[k0 + 0] * dy * dy);
            alpha = fminf(0.99f, s_op[k0 + 0] * __expf(e));
            w0 = T * alpha; T *= (1.0f - alpha);

            dx = px - s_mx[k0 + 1]; dy = py - s_my[k0 + 1];
            e = -(s_a[k0 + 1] * dx * dx + s_b2[k0 + 1] * dx * dy + s_c[k0 + 1] * dy * dy);
            alpha = fminf(0.99f, s_op[k0 + 1] * __expf(e));
            w1 = T * alpha; T *= (1.0f - alpha);

            dx = px - s_mx[k0 + 2]; dy = py - s_my[k0 + 2];
            e = -(s_a[k0 + 2] * dx * dx + s_b2[k0 + 2] * dx * dy + s_c[k0 + 2] * dy * dy);
            alpha = fminf(0.99f, s_op[k0 + 2] * __expf(e));
            w2 = T * alpha; T *= (1.0f - alpha);

            dx = px - s_mx[k0 + 3]; dy = py - s_my[k0 + 3];
            e = -(s_a[k0 + 3] * dx * dx + s_b2[k0 + 3] * dx * dy + s_c[k0 + 3] * dy * dy);
            alpha = fminf(0.99f, s_op[k0 + 3] * __expf(e));
            w3 = T * alpha; T *= (1.0f - alpha);
        }

        // A: weights tile (16 pixels x 4 gaussians), half-wave selects K pair.
        v2f A;
        A[0] = hi ? w2 : w0;
        A[1] = hi ? w3 : w1;

        // B: colors tile (4 gaussians x 16 cols); single aligned ds_load_b64.
        // kb = k0 + 2*hi is even -> 8-byte aligned within a 1024B-aligned row.
        const int kb = k0 + 2 * hi;
        const v2f B = *(const v2f*)(colrow + kb);

        // D(16x16,f32) += A(16x4,f32) * B(4x16,f32)
        acc = __builtin_amdgcn_wmma_f32_16x16x4_f32(
            /*neg_a=*/false, A, /*neg_b=*/false, B,
            /*c_mod=*/(short)0, acc, /*reuse_a=*/false, /*reuse_b=*/false);
    }

    // Background term: bg = prod_n (1 - alpha[n]) = final T (identical in both
    // half-waves; lanes 0-15 publish it).
    if (lane < 16) s_T[wave * kPixPerWave + m] = T;

    // Scatter D into LDS staging. C/D layout: VGPR r holds M = 8*hi + r, N = m.
    if (n_col < 3) {
        #pragma unroll
        for (int r = 0; r < 8; ++r) {
            s_out[(wave * kPixPerWave + 8 * hi + r) * 3 + n_col] = acc[r];
        }
    }
    __syncthreads();

    // Coalesced output: rendered + bg, clamped to [0,1].
    const int base = blockIdx.x * kPixPerBlock * 3;
    for (int i = tid; i < kPixPerBlock * 3; i += 256) {
        const float v = s_out[i] + s_T[i / 3];
        out[base + i] = fminf(1.0f, fmaxf(0.0f, v));
    }
}

extern "C" void kernel_launch(void* const* d_in, const int* in_sizes, int n_in,
                              void* d_out, int out_size, void* d_ws, size_t ws_size,
                              hipStream_t stream) {
    const float* coords      = (const float*)d_in[0];
    const float* means       = (const float*)d_in[1];
    const float* log_scales  = (const float*)d_in[2];
    const float* rotations   = (const float*)d_in[3];
    const float* raw_colors  = (const float*)d_in[4];
    const float* raw_opac    = (const float*)d_in[5];
    float* out = (float*)d_out;

    const int n_pix = kH * kW;
    dim3 grid(n_pix / kPixPerBlock);   // 2048 blocks
    dim3 block(256);                   // 8 waves of 32
    gs2d_render_kernel<<<grid, block, 0, stream>>>(
        coords, means, log_scales, rotations, raw_colors, raw_opac, out);
}